// RouteNet_72112500900101
// MI455X (gfx1250) — compile-verified
//
#include <hip/hip_runtime.h>

typedef __attribute__((ext_vector_type(2))) float v2f;
typedef __attribute__((ext_vector_type(8))) float v8f;

#define NLAY   4
#define WB     4
#define DD     10
#define NIN    784
#define NOUT   10
#define BSZ    131072
#define NPAD   48         // 40 output neurons padded to 48
#define KPAIRS 392        // 784 / 2

// ---------------------------------------------------------------------------
// Pack W_in (flattened [40][784], row n = w*10+d) into pair-interleaved
// B-operand form: bpack[p][n] = (W_in[n][2p], W_in[n][2p+1]), n padded to 48.
// Matches WMMA f32 16x16x4 B layout: lane holds (K=2g+0, K=2g+1) at N=lane&15.
// ---------------------------------------------------------------------------
__global__ void prep_bpack(const float* __restrict__ W_in, float* __restrict__ bpack) {
    int idx = blockIdx.x * blockDim.x + threadIdx.x;
    if (idx >= KPAIRS * NPAD) return;
    int p = idx / NPAD, n = idx % NPAD;
    float v0 = 0.f, v1 = 0.f;
    if (n < WB * DD) {
        v0 = W_in[n * NIN + 2 * p];
        v1 = W_in[n * NIN + 2 * p + 1];
    }
    bpack[2 * idx]     = v0;
    bpack[2 * idx + 1] = v1;
}

// ---------------------------------------------------------------------------
// acts0[b][n] = relu(x[b]·W_in[n,:] + b_in[n]) via V_WMMA_F32_16X16X4_F32.
// One wave per 16-row tile, 3 column tiles (48 cols), 196 K-steps.
// ---------------------------------------------------------------------------
__global__ void __launch_bounds__(256)
input_gemm(const float* __restrict__ x, const float* __restrict__ bpack,
           const float* __restrict__ b_in, float* __restrict__ acts0) {
    const int lane = threadIdx.x & 31;
    const int wave = threadIdx.x >> 5;
    const int tile = blockIdx.x * 8 + wave;      // 16-row tile index
    const int row0 = tile * 16;
    const int m = lane & 15;                     // row (A) / col (B,C) within tile
    const int g = lane >> 4;                     // lane group

    // A layout: lane(g,m) holds x[row0+m][4k + 2g + {0,1}]
    const float* arow = x + (size_t)(row0 + m) * NIN + 2 * g;
    const v2f* bp = (const v2f*)bpack;

    v8f c0 = {}; v8f c1 = {}; v8f c2 = {};
    for (int k = 0; k < NIN / 4; ++k) {
        v2f a = *(const v2f*)(arow + 4 * k);
        const int p = 2 * k + g;                 // K-pair index for this lane group
        v2f b0 = bp[p * NPAD + m];
        v2f b1 = bp[p * NPAD + m + 16];
        v2f b2 = bp[p * NPAD + m + 32];
        c0 = __builtin_amdgcn_wmma_f32_16x16x4_f32(false, a, false, b0, (short)0, c0, false, false);
        c1 = __builtin_amdgcn_wmma_f32_16x16x4_f32(false, a, false, b1, (short)0, c1, false, false);
        c2 = __builtin_amdgcn_wmma_f32_16x16x4_f32(false, a, false, b2, (short)0, c2, false, false);
    }

    // C layout: VGPR v -> row v + 8*g, col = lane&15 (per 16-col tile)
    float bias0 = (m      < WB * DD) ? b_in[m]      : 0.f;
    float bias1 = (m + 16 < WB * DD) ? b_in[m + 16] : 0.f;
    float bias2 = 0.f;                           // cols 32..47: only 32..39 real
    if (m + 32 < WB * DD) bias2 = b_in[m + 32];
    float* orow = acts0 + (size_t)(row0 + 8 * g) * NPAD + m;
    #pragma unroll
    for (int v = 0; v < 8; ++v) {
        orow[(size_t)v * NPAD + 0]  = fmaxf(c0[v] + bias0, 0.f);
        orow[(size_t)v * NPAD + 16] = fmaxf(c1[v] + bias1, 0.f);
        orow[(size_t)v * NPAD + 32] = fmaxf(c2[v] + bias2, 0.f);
    }
}

// ---------------------------------------------------------------------------
// Routing layers + output head. One thread per batch row; all edge weights
// staged into LDS with rows padded to 12 floats (48 B) for aligned b128 reads.
// ---------------------------------------------------------------------------
__global__ void __launch_bounds__(256)
route_layers(const float* __restrict__ acts0,
             const float* __restrict__ gate_w,   // [3][4][4][10]
             const float* __restrict__ data_w,   // [3][4][4][10][10]
             const float* __restrict__ data_b,   // [3][4][4][10]
             const float* __restrict__ W_out,    // [4][10][10]
             float* __restrict__ out) {          // [B*10] outputs + [B] total_gate
    __shared__ __attribute__((aligned(16))) float s_gw[3 * 16 * 12];       // 576
    __shared__ __attribute__((aligned(16))) float s_dw[3 * 16 * 10 * 12];  // 5760
    __shared__ __attribute__((aligned(16))) float s_db[3 * 16 * 12];       // 576
    __shared__ __attribute__((aligned(16))) float s_wo[40 * 12];           // 480

    const int tid = threadIdx.x;
    for (int i = tid; i < 3 * 16 * 12; i += 256)      { int e = i % 12, r = i / 12; s_gw[i] = (e < 10) ? gate_w[r * 10 + e] : 0.f; }
    for (int i = tid; i < 3 * 16 * 10 * 12; i += 256) { int e = i % 12, r = i / 12; s_dw[i] = (e < 10) ? data_w[r * 10 + e] : 0.f; }
    for (int i = tid; i < 3 * 16 * 12; i += 256)      { int e = i % 12, r = i / 12; s_db[i] = (e < 10) ? data_b[r * 10 + e] : 0.f; }
    for (int i = tid; i < 40 * 12; i += 256)          { int e = i % 12, r = i / 12; s_wo[i] = (e < 10) ? W_out[r * 10 + e] : 0.f; }
    __syncthreads();

    const int b = blockIdx.x * 256 + tid;

    float acur[40];
    {
        const float4* ap = (const float4*)(acts0 + (size_t)b * NPAD);
        #pragma unroll
        for (int q = 0; q < 10; ++q) {
            float4 v = ap[q];
            acur[4 * q + 0] = v.x; acur[4 * q + 1] = v.y;
            acur[4 * q + 2] = v.z; acur[4 * q + 3] = v.w;
        }
    }

    float tg = 0.f;
    #pragma unroll
    for (int l = 0; l < NLAY - 1; ++l) {
        float na[40];
        #pragma unroll
        for (int i = 0; i < 40; ++i) na[i] = 0.f;
        #pragma unroll
        for (int s = 0; s < 4; ++s) {
            const float* as = &acur[s * 10];
            #pragma unroll
            for (int t = 0; t < 4; ++t) {
                const int blk = (l * 4 + s) * 4 + t;
                // gate = clip(as · gate_w[l,s,t,:], 0, 1)
                float4 g0 = *(const float4*)&s_gw[blk * 12 + 0];
                float4 g1 = *(const float4*)&s_gw[blk * 12 + 4];
                float2 g2 = *(const float2*)&s_gw[blk * 12 + 8];
                float gsum = as[0] * g0.x + as[1] * g0.y + as[2] * g0.z + as[3] * g0.w
                           + as[4] * g1.x + as[5] * g1.y + as[6] * g1.z + as[7] * g1.w
                           + as[8] * g2.x + as[9] * g2.y;
                float gate = fminf(fmaxf(gsum, 0.f), 1.f);
                tg += gate;
                // dv = as @ data_w[l,s,t] + data_b[l,s,t]
                float dv[10];
                const float* dbp = &s_db[blk * 12];
                #pragma unroll
                for (int e = 0; e < 10; ++e) dv[e] = dbp[e];
                #pragma unroll
                for (int i = 0; i < 10; ++i) {
                    const float av = as[i];
                    const float* drow = &s_dw[(blk * 10 + i) * 12];
                    float4 w0 = *(const float4*)(drow + 0);
                    float4 w1 = *(const float4*)(drow + 4);
                    float2 w2 = *(const float2*)(drow + 8);
                    dv[0] += av * w0.x; dv[1] += av * w0.y; dv[2] += av * w0.z; dv[3] += av * w0.w;
                    dv[4] += av * w1.x; dv[5] += av * w1.y; dv[6] += av * w1.z; dv[7] += av * w1.w;
                    dv[8] += av * w2.x; dv[9] += av * w2.y;
                }
                float* nt = &na[t * 10];
                #pragma unroll
                for (int e = 0; e < 10; ++e) nt[e] += gate * dv[e];
            }
        }
        #pragma unroll
        for (int i = 0; i < 40; ++i) acur[i] = fmaxf(na[i], 0.f);
    }

    // output[b][c] = sum_w acur[w]· W_out[w,c,:]
    #pragma unroll
    for (int c = 0; c < NOUT; ++c) {
        float o = 0.f;
        #pragma unroll
        for (int w = 0; w < 4; ++w) {
            const float* as = &acur[w * 10];
            const float* wr = &s_wo[(w * 10 + c) * 12];
            float4 w0 = *(const float4*)(wr + 0);
            float4 w1 = *(const float4*)(wr + 4);
            float2 w2 = *(const float2*)(wr + 8);
            o += as[0] * w0.x + as[1] * w0.y + as[2] * w0.z + as[3] * w0.w
               + as[4] * w1.x + as[5] * w1.y + as[6] * w1.z + as[7] * w1.w
               + as[8] * w2.x + as[9] * w2.y;
        }
        out[(size_t)b * NOUT + c] = o;
    }
    out[(size_t)BSZ * NOUT + b] = tg;
}

extern "C" void kernel_launch(void* const* d_in, const int* in_sizes, int n_in,
                              void* d_out, int out_size, void* d_ws, size_t ws_size,
                              hipStream_t stream) {
    const float* x      = (const float*)d_in[0];
    const float* W_in   = (const float*)d_in[1];
    const float* b_in   = (const float*)d_in[2];
    const float* gate_w = (const float*)d_in[3];
    const float* data_w = (const float*)d_in[4];
    const float* data_b = (const float*)d_in[5];
    const float* W_out  = (const float*)d_in[6];
    float* out = (float*)d_out;

    float* acts0 = (float*)d_ws;                                   // BSZ*48 floats
    float* bpack = (float*)((char*)d_ws + (size_t)BSZ * NPAD * sizeof(float));

    prep_bpack<<<(KPAIRS * NPAD + 255) / 256, 256, 0, stream>>>(W_in, bpack);
    input_gemm<<<(BSZ / 16) / 8, 256, 0, stream>>>(x, bpack, b_in, acts0);
    route_layers<<<BSZ / 256, 256, 0, stream>>>(acts0, gate_w, data_w, data_b, W_out, out);
}